// CenterNet_Resnet50_30262339568132
// MI455X (gfx1250) — compile-verified
//
#include <hip/hip_runtime.h>
#include <math.h>

namespace {
constexpr int kB = 32, kC = 80, kH = 256, kW = 256;
constexpr int kHW = kH * kW;
constexpr int TW = 32, TH = 16;                // spatial tile per block (512 px)
constexpr int HALO_W = TW + 2;                 // 34
constexpr int HALO_H = TH + 2;                 // 18
constexpr int HALO_N = HALO_W * HALO_H;        // 612 elements per channel tile
constexpr int NBUF = 3;                        // triple buffer -> 1 barrier/chan
constexpr float kConfThresh = 0.3f;
}

// gfx1250 async global->LDS DMA (ASYNCcnt-tracked). GVS mode:
// mem_addr = SGPR64 + VGPR32_byte_offset ; LDS dest byte addr from a VGPR.
__device__ __forceinline__ void async_tile_b32(const float* sbase,
                                               unsigned lds_byte_addr,
                                               unsigned glb_byte_off) {
  asm volatile("global_load_async_to_lds_b32 %0, %1, %2"
               :
               : "v"(lds_byte_addr), "v"(glb_byte_off), "s"(sbase)
               : "memory");
}

__global__ __launch_bounds__(256) void centernet_decode(
    const float* __restrict__ hm, const float* __restrict__ wh,
    const float* __restrict__ reg, float* __restrict__ out) {
  // triple-buffered halo tile of raw heatmap logits (3 * 2448 B = 7.3 KB)
  __shared__ float buf[NBUF][HALO_H][HALO_W];

  const int tid = threadIdx.x;
  const int blk = blockIdx.x;
  const int b  = blk >> 7;          // 128 tiles per batch image
  const int t  = blk & 127;
  const int y0 = (t >> 3) * TH;     // 16 tile rows
  const int x0 = (t & 7) * TW;      // 8 tile cols

  // --- halo-load element assignment: 3 async rounds, full EXEC each round
  // (remainder lanes duplicate-write identical data; keeps ASYNCcnt uniform)
  const int e0 = tid;                              // 0..255
  const int e1 = 256 + tid;                        // 256..511
  const int e2 = 512 + (tid % (HALO_N - 512));     // 512..611 (dups ok)
  const int hy0 = e0 / HALO_W, hx0 = e0 - hy0 * HALO_W;
  const int hy1 = e1 / HALO_W, hx1 = e1 - hy1 * HALO_W;
  const int hy2 = e2 / HALO_W, hx2 = e2 - hy2 * HALO_W;
  // replicate-clamp == -inf SAME padding for a 3x3 max (clamped neighbor
  // duplicates an element already inside the window)
  const int gy0 = min(max(y0 - 1 + hy0, 0), kH - 1);
  const int gx0 = min(max(x0 - 1 + hx0, 0), kW - 1);
  const int gy1 = min(max(y0 - 1 + hy1, 0), kH - 1);
  const int gx1 = min(max(x0 - 1 + hx1, 0), kW - 1);
  const int gy2 = min(max(y0 - 1 + hy2, 0), kH - 1);
  const int gx2 = min(max(x0 - 1 + hx2, 0), kW - 1);
  const unsigned voff0 = (unsigned)((gy0 * kW + gx0) * 4);
  const unsigned voff1 = (unsigned)((gy1 * kW + gx1) * 4);
  const unsigned voff2 = (unsigned)((gy2 * kW + gx2) * 4);

  // low 32 bits of a flat shared pointer == LDS byte address (aperture layout)
  const unsigned lbase = (unsigned)(uintptr_t)(&buf[0][0][0]);
  const unsigned l0 = lbase + (unsigned)(e0 * 4);
  const unsigned l1 = lbase + (unsigned)(e1 * 4);
  const unsigned l2 = lbase + (unsigned)(e2 * 4);
  const unsigned bufStrideB = (unsigned)(HALO_N * 4);  // 2448 B

  const float* plane = hm + (size_t)b * kC * kHW;

  // prologue: kick off channel 0 into buffer 0
  async_tile_b32(plane, l0, voff0);
  async_tile_b32(plane, l1, voff1);
  async_tile_b32(plane, l2, voff2);

  const int wy = tid >> 5;   // 0..7 -> pixel rows 2*wy, 2*wy+1
  const int lx = tid & 31;   // 0..31

  float best0 = -INFINITY, best1 = -INFINITY;
  int bestC0 = 0, bestC1 = 0;

#pragma unroll 1
  for (int c = 0; c < kC; ++c) {
    if (c + 1 < kC) {
      // overlap: DMA channel c+1 while computing channel c
      const float* np = plane + (size_t)(c + 1) * kHW;
      const unsigned nb = (unsigned)((c + 1) % NBUF) * bufStrideB;
      async_tile_b32(np, l0 + nb, voff0);
      async_tile_b32(np, l1 + nb, voff1);
      async_tile_b32(np, l2 + nb, voff2);
      // 3 newest (c+1) may stay in flight; channel c's trio has retired
      asm volatile("s_wait_asynccnt 3" ::: "memory");
    } else {
      asm volatile("s_wait_asynccnt 0" ::: "memory");
    }
    // single barrier per channel: also separates compute(c-2) reads from the
    // issue(c+1) writes into buffer (c+1)%3 == (c-2)%3 (triple buffering)
    __syncthreads();

    const float(*tile)[HALO_W] = buf[c % NBUF];
    const int r = 2 * wy;  // halo row of the upper pixel's top window row
    // 4 window rows cover both vertically-adjacent pixels (12 LDS reads)
    const float r0a = tile[r + 0][lx], r0b = tile[r + 0][lx + 1], r0c = tile[r + 0][lx + 2];
    const float r1a = tile[r + 1][lx], r1b = tile[r + 1][lx + 1], r1c = tile[r + 1][lx + 2];
    const float r2a = tile[r + 2][lx], r2b = tile[r + 2][lx + 1], r2c = tile[r + 2][lx + 2];
    const float r3a = tile[r + 3][lx], r3b = tile[r + 3][lx + 1], r3c = tile[r + 3][lx + 2];
    const float m0 = fmaxf(fmaxf(r0a, r0b), r0c);
    const float m1 = fmaxf(fmaxf(r1a, r1b), r1c);
    const float m2 = fmaxf(fmaxf(r2a, r2b), r2c);
    const float m3 = fmaxf(fmaxf(r3a, r3b), r3c);
    const float wmax0 = fmaxf(fmaxf(m0, m1), m2);
    const float wmax1 = fmaxf(fmaxf(m1, m2), m3);
    // pool-NMS peak test + running channel argmax, in logit space (sigmoid
    // strictly monotone); strict '>' keeps first-occurrence tie semantics
    if (r1b == wmax0 && r1b > best0) { best0 = r1b; bestC0 = c; }
    if (r2b == wmax1 && r2b > best1) { best1 = r2b; bestC1 = c; }
  }

  // --- decode boxes for the two pixels this thread owns ---
  const int gx = x0 + lx;
  const float inv = 1.0f / 256.0f;  // W == H == 256

#pragma unroll
  for (int k = 0; k < 2; ++k) {
    const int gy = y0 + 2 * wy + k;
    const int p = gy * kW + gx;
    const float bl = (k == 0) ? best0 : best1;
    const int bc = (k == 0) ? bestC0 : bestC1;
    const float conf = 1.0f / (1.0f + __expf(-bl));  // 0 when no peak

    float ox1 = 0.f, oy1 = 0.f, ox2 = 0.f, oy2 = 0.f, ocf = 0.f, ocp = 0.f;
    if (conf > kConfThresh) {
      const size_t pb = (size_t)b * 2 * kHW + (size_t)p;
      const float r0 = reg[pb], r1 = reg[pb + kHW];
      const float w0 = wh[pb], w1 = wh[pb + kHW];
      const float cx = (float)gx + r0;
      const float cy = (float)gy + r1;
      const float hw = w0 * 0.5f;
      const float hh = w1 * 0.5f;
      ox1 = (cx - hw) * inv;
      oy1 = (cy - hh) * inv;
      ox2 = (cx + hw) * inv;
      oy2 = (cy + hh) * inv;
      ocf = conf;
      ocp = (float)bc;
    }
    // dets[b, p, 0:6]; 24B rows are 8B aligned -> three b64 stores
    float2* o = (float2*)(out + ((size_t)b * kHW + (size_t)p) * 6);
    o[0] = make_float2(ox1, oy1);
    o[1] = make_float2(ox2, oy2);
    o[2] = make_float2(ocf, ocp);
  }
}

extern "C" void kernel_launch(void* const* d_in, const int* in_sizes, int n_in,
                              void* d_out, int out_size, void* d_ws,
                              size_t ws_size, hipStream_t stream) {
  const float* hm = (const float*)d_in[0];
  const float* wh = (const float*)d_in[1];
  const float* reg = (const float*)d_in[2];
  float* out = (float*)d_out;
  // 32 batches * (256x256 plane tiled as 16x8 blocks of 32x16 px) = 4096
  const int blocks = kB * (kH / TH) * (kW / TW);
  centernet_decode<<<blocks, 256, 0, stream>>>(hm, wh, reg, out);
}